// SpaceAttentionModule_9680856285922
// MI455X (gfx1250) — compile-verified
//
#include <hip/hip_runtime.h>
#include <hip/hip_bf16.h>

typedef float v2f __attribute__((ext_vector_type(2)));
typedef float v8f __attribute__((ext_vector_type(8)));

#define ALPHA 0.2f
// B=16, K=256, T=64, D=128, E=2D=256

// ---------------------------------------------------------------------------
// Kernel 1: uv = x @ Wcat^T   (M=4096, N=256, Kdep=64), f32 WMMA 16x16x4
//   Wcat[e,t] = W[e&127, ((e>>7)<<6) + t]   (folds W1/W2 split)
// One wave (32 threads) per 16x16 tile, 16 chained WMMAs over K.
// ---------------------------------------------------------------------------
__global__ __launch_bounds__(32) void uv_gemm(const float* __restrict__ x,
                                              const float* __restrict__ W,
                                              float* __restrict__ uv) {
  const int tile  = blockIdx.x;      // 0..4095 (256 mtiles x 16 ntiles)
  const int mtile = tile >> 4;
  const int ntile = tile & 15;
  const int lane  = threadIdx.x;
  const int mloc  = lane & 15;
  const int half  = lane >> 4;       // K-pair selector per ISA A/B layout

  const int row = mtile * 16 + mloc;                 // flattened (b*K + k)
  const int e   = ntile * 16 + mloc;                 // output column (N = lane&15)
  const float* __restrict__ xrow = x + row * 64;
  const float* __restrict__ wrow = W + (e & 127) * 128 + ((e >> 7) << 6);

  v8f c = {};
#pragma unroll
  for (int kk = 0; kk < 16; ++kk) {
    const int t0 = kk * 4 + half * 2;
    v2f aF, bF;
    aF.x = xrow[t0];  aF.y = xrow[t0 + 1];           // A[M=mloc, K=t0..t0+1]
    bF.x = wrow[t0];  bF.y = wrow[t0 + 1];           // B[K=t0..t0+1, N=mloc]
    c = __builtin_amdgcn_wmma_f32_16x16x4_f32(false, aF, false, bF,
                                              (short)0, c, false, false);
  }

  const int n = ntile * 16 + mloc;
#pragma unroll
  for (int g = 0; g < 8; ++g) {
    const int m = mtile * 16 + g + half * 8;         // C layout: M = g + 8*half
    uv[m * 256 + n] = c[g];
  }
}

// ---------------------------------------------------------------------------
// Kernel 2: fused e-tile + bias + softmax -> attention
// Block = (b, 16-row i-tile), 256 threads (8 waves). LDS ~158KB of 320KB WGP.
//   u_s : 16x128   (u + b_lin folded)
//   a_s : 128
//   v_s : 256x130  (pad 130 -> bank = (2*lane + d) % 64, conflict-free)
//   e_s : 16x257   (softmax staging)
// ---------------------------------------------------------------------------
__global__ __launch_bounds__(256) void attn_rows(const float* __restrict__ uv,
                                                 const float* __restrict__ b_lin,
                                                 const float* __restrict__ a,
                                                 const float* __restrict__ bias,
                                                 float* __restrict__ att) {
  extern __shared__ float sm[];
  float* u_s = sm;                  // 2048
  float* a_s = u_s + 16 * 128;      // 128
  float* v_s = a_s + 128;           // 33280
  float* e_s = v_s + 256 * 130;     // 4112

  const int tid = threadIdx.x;
  const int b   = blockIdx.x >> 4;
  const int i0  = (blockIdx.x & 15) << 4;
  const float* __restrict__ uvb = uv + b * 256 * 256;

  for (int idx = tid; idx < 16 * 128; idx += 256) {
    const int ii = idx >> 7, d = idx & 127;
    u_s[idx] = uvb[(i0 + ii) * 256 + d] + b_lin[d];
  }
  if (tid < 128) a_s[tid] = a[tid];
  for (int idx = tid; idx < 256 * 128; idx += 256) {   // coalesced v stage
    const int j = idx >> 7, d = idx & 127;
    v_s[j * 130 + d] = uvb[j * 256 + 128 + d];
  }
  __syncthreads();

  const int j = tid;
  float acc[16];
#pragma unroll
  for (int ii = 0; ii < 16; ++ii) acc[ii] = 0.f;

  for (int d = 0; d < 128; ++d) {
    const float vd = v_s[j * 130 + d];     // conflict-free (stride 130)
    const float ad = a_s[d];               // broadcast
#pragma unroll
    for (int ii = 0; ii < 16; ++ii) {
      const float z = u_s[ii * 128 + d] + vd;   // broadcast + add
      const float l = (z >= 0.f) ? z : ALPHA * z;
      acc[ii] = fmaf(l, ad, acc[ii]);
    }
  }
#pragma unroll
  for (int ii = 0; ii < 16; ++ii)
    e_s[ii * 257 + j] = acc[ii] + bias[(i0 + ii) * 256 + j];
  __syncthreads();

  // softmax over j: 8 waves x 2 rows, wave32 shuffle reductions
  const int wave = tid >> 5, lane = tid & 31;
#pragma unroll
  for (int rr = 0; rr < 2; ++rr) {
    const int r = wave + rr * 8;
    float* row = e_s + r * 257;
    float m = -1e30f;
#pragma unroll
    for (int s = 0; s < 8; ++s) m = fmaxf(m, row[lane + 32 * s]);
#pragma unroll
    for (int off = 16; off > 0; off >>= 1) m = fmaxf(m, __shfl_xor(m, off, 32));
    float ex[8];
    float sum = 0.f;
#pragma unroll
    for (int s = 0; s < 8; ++s) { ex[s] = __expf(row[lane + 32 * s] - m); sum += ex[s]; }
#pragma unroll
    for (int off = 16; off > 0; off >>= 1) sum += __shfl_xor(sum, off, 32);
    const float inv = 1.f / sum;
    float* __restrict__ arow = att + (b * 256 + i0 + r) * 256;
#pragma unroll
    for (int s = 0; s < 8; ++s) arow[lane + 32 * s] = ex[s] * inv;
  }
}

// ---------------------------------------------------------------------------
// Kernel 3: h = sigmoid(att @ x) per batch  (256x256 @ 256x64), f32 WMMA
// One wave per 16x16 tile, 64 chained WMMAs over Kdep=256, fused sigmoid.
// ---------------------------------------------------------------------------
__global__ __launch_bounds__(32) void out_gemm(const float* __restrict__ att,
                                               const float* __restrict__ x,
                                               float* __restrict__ out) {
  const int blk   = blockIdx.x;      // 0..1023 (16 b x 16 mtiles x 4 ntiles)
  const int b     = blk >> 6;
  const int rem   = blk & 63;
  const int mtile = rem >> 2;
  const int ntile = rem & 3;
  const int lane  = threadIdx.x;
  const int mloc  = lane & 15;
  const int half  = lane >> 4;

  const float* __restrict__ arow = att + (b * 256 + mtile * 16 + mloc) * 256;
  const float* __restrict__ bx   = x + b * 256 * 64;
  const int n = ntile * 16 + mloc;

  v8f c = {};
#pragma unroll 8
  for (int kk = 0; kk < 64; ++kk) {
    const int t0 = kk * 4 + half * 2;
    v2f aF, bF;
    aF.x = arow[t0];          aF.y = arow[t0 + 1];
    bF.x = bx[t0 * 64 + n];   bF.y = bx[(t0 + 1) * 64 + n];
    c = __builtin_amdgcn_wmma_f32_16x16x4_f32(false, aF, false, bF,
                                              (short)0, c, false, false);
  }
#pragma unroll
  for (int g = 0; g < 8; ++g) {
    const int m = mtile * 16 + g + half * 8;
    const float vv = c[g];
    out[(b * 256 + m) * 64 + n] = 1.f / (1.f + __expf(-vv));
  }
}

// ---------------------------------------------------------------------------
extern "C" void kernel_launch(void* const* d_in, const int* in_sizes, int n_in,
                              void* d_out, int out_size, void* d_ws, size_t ws_size,
                              hipStream_t stream) {
  const float* x     = (const float*)d_in[0];
  // d_in[1] = embedding: unused by the reference computation
  const float* W     = (const float*)d_in[2];
  const float* b_lin = (const float*)d_in[3];
  const float* a     = (const float*)d_in[4];
  const float* bias  = (const float*)d_in[5];
  float* out = (float*)d_out;

  float* uv  = (float*)d_ws;                 // 16*256*256 f32 = 4 MB
  float* att = uv + 16 * 256 * 256;          // 16*256*256 f32 = 4 MB

  uv_gemm<<<4096, 32, 0, stream>>>(x, W, uv);

  const size_t lds_bytes =
      (16 * 128 + 128 + 256 * 130 + 16 * 257) * sizeof(float);  // ~158 KB
  attn_rows<<<16 * 16, 256, lds_bytes, stream>>>(uv, b_lin, a, bias, att);

  out_gemm<<<16 * 64, 32, 0, stream>>>(att, x, out);
}